// E3nnMLPS2Grid_63024350102053
// MI455X (gfx1250) — compile-verified
//
#include <hip/hip_runtime.h>
#include <math.h>

#define EPSV 1e-5f
#define INV_SQRT128 0.08838834764831845f
#define INV_SQRT2   0.7071067811865476f
#define B_TOTAL 8192
#define GGRID 10000

typedef __attribute__((ext_vector_type(2))) float v2f;
typedef __attribute__((ext_vector_type(8))) float v8f;

__device__ __forceinline__ v8f wmma_f32x4(v2f a, v2f b, v8f c) {
    // D = A(16x4) * B(4x16) + C(16x16), fp32 matrix pipe
    return __builtin_amdgcn_wmma_f32_16x16x4_f32(false, a, false, b, (short)0, c, false, false);
}

// async copy 8 bytes global -> LDS (per active lane), tracked by ASYNCcnt
__device__ __forceinline__ void async_b64(unsigned int lds_off, const float* gaddr) {
    asm volatile("global_load_async_to_lds_b64 %0, %1, off"
                 :: "v"(lds_off), "v"(gaddr) : "memory");
}

// ---------------- Stage-0 batch moments of x: m11,m12,m22 ----------------
__global__ __launch_bounds__(256) void k_stats0(const float* __restrict__ x,
                                                float* __restrict__ consts) {
    float s11 = 0.f, s12 = 0.f, s22 = 0.f;
    for (int b = threadIdx.x; b < B_TOTAL; b += 256) {
        const float* r = x + b * 6;
#pragma unroll
        for (int d = 0; d < 3; ++d) {
            float a = r[d], c = r[3 + d];
            s11 += a * a; s12 += a * c; s22 += c * c;
        }
    }
    __shared__ float sm[256];
    float vals[3] = {s11, s12, s22};
    for (int q = 0; q < 3; ++q) {
        sm[threadIdx.x] = vals[q]; __syncthreads();
        for (int s = 128; s > 0; s >>= 1) {
            if (threadIdx.x < s) sm[threadIdx.x] += sm[threadIdx.x + s];
            __syncthreads();
        }
        if (threadIdx.x == 0) consts[q] = sm[0] / (B_TOTAL * 3.0f);
        __syncthreads();
    }
}

// ------------- Block-0 coefficients alpha/beta/z0c + output K -------------
__global__ __launch_bounds__(128) void k_prep0(const float* __restrict__ W1_in,
                                               const float* __restrict__ W2,
                                               const float* __restrict__ bn_w,
                                               const float* __restrict__ bn_b,
                                               const float* __restrict__ Wout,
                                               const float* __restrict__ W3,
                                               float* consts, float* scaleA, float* biasA) {
    int w = threadIdx.x;
    float m11 = consts[0], m12 = consts[1], m22 = consts[2];
    float a  = W1_in[w];         // u=0
    float bb = W1_in[128 + w];   // u=1
    float norm = 0.5f * (a * a * m11 + 2.f * a * bb * m12 + bb * bb * m22);
    float sc = rsqrtf(norm + EPSV) * bn_w[(0 * 6 + 1) * 128 + w];
    float w2 = W2[(0 * 6 + 1) * 128 + w];
    float tA = w2 * a  * INV_SQRT2 * sc * INV_SQRT128;
    float tB = w2 * bb * INV_SQRT2 * sc * INV_SQRT128;
    float tC = W2[(0 * 6 + 0) * 128 + w] * bn_b[w] * INV_SQRT128;       // bn_b[0][w]
    float tK = Wout[w] * W3[(3 * 6 + 1) * 128 + w] * INV_SQRT128;
    __shared__ float sm[128];
    float vals[4] = {tA, tB, tC, tK};
    for (int q = 0; q < 4; ++q) {
        sm[w] = vals[q]; __syncthreads();
        for (int s = 64; s > 0; s >>= 1) {
            if (w < s) sm[w] += sm[w + s];
            __syncthreads();
        }
        if (w == 0) consts[3 + q] = sm[0];
        __syncthreads();
    }
    if (w < 36) { scaleA[w] = 1.0f; biasA[w] = 0.0f; }   // identity affine for blk 0
}

// ---------------- z init for block 0 (post-BN, post lin_down) -------------
__global__ __launch_bounds__(256) void k_zinit(const float* __restrict__ x,
                                               const float* __restrict__ consts,
                                               float* __restrict__ z) {
    int b = blockIdx.x * 256 + threadIdx.x;
    if (b >= B_TOTAL) return;
    float alpha = consts[3], beta = consts[4], z0c = consts[5];
    float* r = z + b * 36;
    r[0] = z0c;
#pragma unroll
    for (int d = 0; d < 3; ++d) r[1 + d] = alpha * x[b * 6 + d] + beta * x[b * 6 + 3 + d];
#pragma unroll
    for (int k = 4; k < 36; ++k) r[k] = 0.f;
}

// ----- per-l channel vector v[l] = W1_blocks[blk][l]^T W3[blk][l] /sqrt128 -----
__global__ __launch_bounds__(128) void k_v(const float* __restrict__ W1_blocks,
                                           const float* __restrict__ W3,
                                           int blk, float* __restrict__ v) {
    int l = blockIdx.x, w = threadIdx.x;
    const float* Wm = W1_blocks + ((size_t)blk * 6 + l) * 16384;
    const float* w3 = W3 + (blk * 6 + l) * 128;
    float s = 0.f;
    for (int u = 0; u < 128; ++u) s += Wm[u * 128 + w] * w3[u];
    v[l * 128 + w] = s * INV_SQRT128;
}

// --------- batch stats over z[8192,36] + fold BN+lin_down into affine ---------
__global__ __launch_bounds__(1024) void k_stats(const float* __restrict__ z,
                                                const float* __restrict__ v,
                                                const float* __restrict__ W2,
                                                const float* __restrict__ bn_w,
                                                const float* __restrict__ bn_b,
                                                int blk, float* scaleA, float* biasA) {
    __shared__ float sm[1024];
    __shared__ float red[8];
    __shared__ float segs[6];
    int tid = threadIdx.x;
    float p[7] = {0, 0, 0, 0, 0, 0, 0};
    for (int b = tid; b < B_TOTAL; b += 1024) {
        const float* r = z + b * 36;
        float z0 = r[0];
        p[0] += z0; p[1] += z0 * z0;
        float s;
        s = 0; for (int k = 1;  k < 4;  ++k) s += r[k] * r[k]; p[2] += s;
        s = 0; for (int k = 4;  k < 9;  ++k) s += r[k] * r[k]; p[3] += s;
        s = 0; for (int k = 9;  k < 16; ++k) s += r[k] * r[k]; p[4] += s;
        s = 0; for (int k = 16; k < 25; ++k) s += r[k] * r[k]; p[5] += s;
        s = 0; for (int k = 25; k < 36; ++k) s += r[k] * r[k]; p[6] += s;
    }
    for (int q = 0; q < 7; ++q) {
        sm[tid] = p[q]; __syncthreads();
        for (int s = 512; s > 0; s >>= 1) {
            if (tid < s) sm[tid] += sm[tid + s];
            __syncthreads();
        }
        if (tid == 0) red[q] = sm[0];
        __syncthreads();
    }
    float zbar = red[0] / (float)B_TOTAL;
    float var0 = red[1] / (float)B_TOTAL - zbar * zbar;
    int l = tid >> 7, w = tid & 127;
    float term = 0.f;
    if (l < 6) {
        float M = (l == 0) ? var0 : red[1 + l] / ((float)B_TOTAL * (2.f * l + 1.f));
        float vv = v[l * 128 + w];
        term = W2[(blk * 6 + l) * 128 + w] * vv * bn_w[(blk * 6 + l) * 128 + w]
               * rsqrtf(vv * vv * M + EPSV) * INV_SQRT128;
    }
    sm[tid] = term; __syncthreads();
    for (int s = 64; s > 0; s >>= 1) {            // segmented reduce per l
        if (w < s) sm[tid] += sm[tid + s];
        __syncthreads();
    }
    if (w == 0 && l < 6) segs[l] = sm[tid];
    __syncthreads();
    float ct = (tid < 128) ? W2[(blk * 6 + 0) * 128 + tid] * bn_b[blk * 128 + tid] * INV_SQRT128 : 0.f;
    sm[tid] = ct; __syncthreads();
    for (int s = 64; s > 0; s >>= 1) {
        if (tid < s) sm[tid] += sm[tid + s];
        __syncthreads();
    }
    if (tid == 0) {
        float A = segs[0], C = sm[0];
        scaleA[0] = A; biasA[0] = C - A * zbar;
        const int offs[7] = {0, 1, 4, 9, 16, 25, 36};
        for (int ll = 1; ll < 6; ++ll)
            for (int k = offs[ll]; k < offs[ll + 1]; ++k) { scaleA[k] = segs[ll]; biasA[k] = 0.f; }
    }
}

// ------------- The heavy kernel: z' = sigmoid(z*Y^T)*Yinv^T (fused) -------------
// 64 blocks x 8 waves; each wave owns a 16-row tile of z. All waves share the same
// G-chunk sequence, so wave 0 async-copies each chunk's Y/Yinv tiles into
// double-buffered LDS (GLOBAL_LOAD_ASYNC_TO_LDS_B64 + s_wait_asynccnt), cutting
// L2 traffic 8x and overlapping copy with WMMA compute.
__global__ __launch_bounds__(256) void k_s2act(const float* __restrict__ zin,
                                               float* __restrict__ zout,
                                               const float* __restrict__ scaleA,
                                               const float* __restrict__ biasA,
                                               const float* __restrict__ Y,
                                               const float* __restrict__ Yinv) {
    const int lane = threadIdx.x & 31;
    const int wave = threadIdx.x >> 5;
    const int m0   = (blockIdx.x * 8 + wave) * 16;
    const int hi   = lane >> 4;      // half-wave
    const int lm   = lane & 15;

    __shared__ float stageY[2][576];     // [buf][g_row(16) * 36]
    __shared__ float stageI[2][576];     // [buf][n(36) * 16 g-cols]
    __shared__ float gT[8][16][16];      // per-wave transpose buffer
    float (*gbuf)[16] = gT[wave];

    // ---- loop-invariant async-copy addressing (wave 0 only uses these) ----
    // Y chunk: 288 float-pairs; pair p: row r=p/18, col off=(p%18)*2
    // I chunk: 288 float-pairs; pair p: n=p>>3, goff=(p&7)*2
    int rY[9], oY[9], nI[9], oI[9];
    unsigned int ldsY[9], ldsI[9];
    const unsigned int baseY = (unsigned int)(size_t)&stageY[0][0];
    const unsigned int baseI = (unsigned int)(size_t)&stageI[0][0];
#pragma unroll
    for (int i = 0; i < 9; ++i) {
        int p = i * 32 + lane;
        rY[i] = p / 18;           oY[i] = (p % 18) * 2;
        nI[i] = p >> 3;           oI[i] = (p & 7) * 2;
        ldsY[i] = baseY + (unsigned int)(rY[i] * 36 + oY[i]) * 4u;
        ldsI[i] = baseI + (unsigned int)(nI[i] * 16 + oI[i]) * 4u;
    }

    // A-matrix rows (z tile) in registers, BN/lin_down affine applied on load.
    // ISA f32 A 16x4 layout: lane<16: v0=K+0,v1=K+1 ; lane>=16: v0=K+2,v1=K+3
    v2f za[9];
    {
        const float* zr = zin + (size_t)(m0 + lm) * 36;
#pragma unroll
        for (int j = 0; j < 9; ++j) {
            int k = 4 * j + 2 * hi;
            za[j].x = zr[k]     * scaleA[k]     + biasA[k];
            za[j].y = zr[k + 1] * scaleA[k + 1] + biasA[k + 1];
        }
    }

    const v8f vzero = {0.f, 0.f, 0.f, 0.f, 0.f, 0.f, 0.f, 0.f};
    v8f acc0 = vzero, acc1 = vzero, acc2 = vzero;

    // prologue: stage chunk 0 into buffer 0 (18 async instructions per chunk)
    if (wave == 0) {
#pragma unroll
        for (int i = 0; i < 9; ++i) {
            async_b64(ldsY[i], Y + (size_t)(0 + rY[i]) * 36 + oY[i]);
            async_b64(ldsI[i], Yinv + (size_t)nI[i] * GGRID + 0 + oI[i]);
        }
    }

    int c = 0;
    for (int g0 = 0; g0 < GGRID; g0 += 16, ++c) {
        const int buf = c & 1;
        if (wave == 0) {
            if (g0 + 16 < GGRID) {
                const int gn = g0 + 16;
                const unsigned int bo = (unsigned int)(buf ^ 1) * 2304u;
#pragma unroll
                for (int i = 0; i < 9; ++i) {
                    async_b64(ldsY[i] + bo, Y + (size_t)(gn + rY[i]) * 36 + oY[i]);
                    async_b64(ldsI[i] + bo, Yinv + (size_t)nI[i] * GGRID + gn + oI[i]);
                }
                // previous chunk's 18 transfers complete (in-order) when <=18 remain
                asm volatile("s_wait_asynccnt 0x12" ::: "memory");
            } else {
                asm volatile("s_wait_asynccnt 0x0" ::: "memory");
            }
        }
        __syncthreads();    // staged data visible to all waves

        const float* sy = &stageY[buf][0];
        const float* si = &stageI[buf][0];

        // GEMM1: d = z_tile[16x36] * Ychunk[36x16]   (B from LDS)
        v8f d = vzero;
#pragma unroll
        for (int j = 0; j < 9; ++j) {
            v2f b;
            b.x = sy[lm * 36 + 4 * j + 2 * hi];
            b.y = sy[lm * 36 + 4 * j + 2 * hi + 1];
            d = wmma_f32x4(za[j], b, d);
        }
        // sigmoid (native exp + rcp)
#pragma unroll
        for (int r = 0; r < 8; ++r)
            d[r] = __builtin_amdgcn_rcpf(1.0f + __expf(-d[r]));
        // transpose g: C-layout (vgpr=row, lane=col) -> A-layout, via per-wave LDS
#pragma unroll
        for (int r = 0; r < 8; ++r) gbuf[r + 8 * hi][lm] = d[r];
        __builtin_amdgcn_wave_barrier();
        v2f ga[4];
#pragma unroll
        for (int j = 0; j < 4; ++j) {
            ga[j].x = gbuf[lm][4 * j + 2 * hi];
            ga[j].y = gbuf[lm][4 * j + 2 * hi + 1];
        }
        __builtin_amdgcn_wave_barrier();
        // GEMM2: acc[16x48] += g[16x16] * YinvT_chunk[16x48]  (cols 36..47 masked)
#pragma unroll
        for (int t = 0; t < 3; ++t) {
            int n = 16 * t + lm;
            float msk = (n < 36) ? 1.0f : 0.0f;
            int nc = (n < 36) ? n : 35;
            v8f* acc = (t == 0) ? &acc0 : (t == 1) ? &acc1 : &acc2;
#pragma unroll
            for (int j = 0; j < 4; ++j) {
                v2f b;
                b.x = si[nc * 16 + 4 * j + 2 * hi] * msk;
                b.y = si[nc * 16 + 4 * j + 2 * hi + 1] * msk;
                *acc = wmma_f32x4(ga[j], b, *acc);
            }
        }
        __syncthreads();    // all waves done with buf before it is overwritten
    }
    // store: C-layout -> zout[row,col], drop padded cols >= 36
#pragma unroll
    for (int t = 0; t < 3; ++t) {
        v8f acc = (t == 0) ? acc0 : (t == 1) ? acc1 : acc2;
        int col = 16 * t + lm;
#pragma unroll
        for (int r = 0; r < 8; ++r) {
            if (col < 36) zout[(size_t)(m0 + r + 8 * hi) * 36 + col] = acc[r];
        }
    }
}

// -------------------------- final 1e projection --------------------------
__global__ __launch_bounds__(256) void k_final(const float* __restrict__ z,
                                               const float* __restrict__ consts,
                                               float* __restrict__ out) {
    int i = blockIdx.x * 256 + threadIdx.x;
    if (i >= B_TOTAL * 3) return;
    int b = i / 3, d = i - b * 3;
    out[i] = consts[6] * z[b * 36 + 1 + d];
}

extern "C" void kernel_launch(void* const* d_in, const int* in_sizes, int n_in,
                              void* d_out, int out_size, void* d_ws, size_t ws_size,
                              hipStream_t stream) {
    const float* x      = (const float*)d_in[0];
    const float* W1_in  = (const float*)d_in[1];
    const float* W1_blk = (const float*)d_in[2];
    const float* bn_w   = (const float*)d_in[3];
    const float* bn_b   = (const float*)d_in[4];
    const float* W2     = (const float*)d_in[5];
    const float* W3     = (const float*)d_in[6];
    const float* Wout   = (const float*)d_in[7];
    const float* Y      = (const float*)d_in[8];
    const float* Yinv   = (const float*)d_in[9];
    (void)in_sizes; (void)n_in; (void)out_size; (void)ws_size;

    float* ws     = (float*)d_ws;
    float* consts = ws;                 // 16 floats
    float* scaleA = ws + 16;            // 64
    float* biasA  = ws + 80;            // 64
    float* vbuf   = ws + 160;           // 768 + pad
    float* zA     = ws + 1024;          // 8192*36
    float* zB     = zA + (size_t)B_TOTAL * 36;

    k_stats0<<<1, 256, 0, stream>>>(x, consts);
    k_prep0 <<<1, 128, 0, stream>>>(W1_in, W2, bn_w, bn_b, Wout, W3, consts, scaleA, biasA);
    k_zinit <<<32, 256, 0, stream>>>(x, consts, zA);

    float* zin = zA; float* zout = zB;
    for (int blk = 0; blk < 4; ++blk) {
        k_s2act<<<64, 256, 0, stream>>>(zin, zout, scaleA, biasA, Y, Yinv);
        if (blk < 3) {
            k_v    <<<6, 128, 0, stream>>>(W1_blk, W3, blk, vbuf);
            k_stats<<<1, 1024, 0, stream>>>(zout, vbuf, W2, bn_w, bn_b, blk + 1, scaleA, biasA);
        }
        float* t = zin; zin = zout; zout = t;
    }
    k_final<<<96, 256, 0, stream>>>(zin, consts, (float*)d_out);
}